// TAG_6399501271543
// MI455X (gfx1250) — compile-verified
//
#include <hip/hip_runtime.h>
#include <cstdint>

typedef float v2f __attribute__((ext_vector_type(2)));
typedef float v4f __attribute__((ext_vector_type(4)));
typedef float v8f __attribute__((ext_vector_type(8)));

typedef long long ll;

// ---------------- utility ----------------
__global__ void k_zero(float* __restrict__ p, ll n) {
  ll t = (ll)blockIdx.x * blockDim.x + threadIdx.x;
  if (t < n) p[t] = 0.f;
}

// deg[col[e]] += ew[e]
__global__ void k_deg(const ll* __restrict__ col, const float* __restrict__ ew,
                      float* __restrict__ deg, ll E) {
  ll e = (ll)blockIdx.x * blockDim.x + threadIdx.x;
  if (e < E) atomicAdd(&deg[col[e]], ew[e]);
}

// in-place: deg -> deg^-0.5 (0 if deg<=0)
__global__ void k_dinv(float* __restrict__ deg, ll n) {
  ll t = (ll)blockIdx.x * blockDim.x + threadIdx.x;
  if (t < n) { float d = deg[t]; deg[t] = (d > 0.f) ? rsqrtf(d) : 0.f; }
}

// norm[e] = dinv[row]*ew*dinv[col]
__global__ void k_norm(const ll* __restrict__ row, const ll* __restrict__ col,
                       const float* __restrict__ ew, const float* __restrict__ dinv,
                       float* __restrict__ norm, ll E) {
  ll e = (ll)blockIdx.x * blockDim.x + threadIdx.x;
  if (e < E) norm[e] = dinv[row[e]] * ew[e] * dinv[col[e]];
}

// ---------------- layer 1 (F_IN = 2) ----------------
// h1 layout: [n][8] = 4 hop slices x 2 features; slice0 = x, rest zero
__global__ void k_init_h1(const float* __restrict__ x, float* __restrict__ h1, ll n) {
  ll t = (ll)blockIdx.x * blockDim.x + threadIdx.x;
  if (t >= n) return;
  float* h = h1 + t * 8;
  h[0] = x[2 * t]; h[1] = x[2 * t + 1];
  h[2] = 0.f; h[3] = 0.f; h[4] = 0.f; h[5] = 0.f; h[6] = 0.f; h[7] = 0.f;
}

__global__ void k_prop2(const ll* __restrict__ row, const ll* __restrict__ col,
                        const float* __restrict__ norm, float* __restrict__ h1,
                        int offin, int offout, ll E) {
  ll e = (ll)blockIdx.x * blockDim.x + threadIdx.x;
  if (e >= E) return;
  ll r = row[e], c = col[e];
  float w = norm[e];
  float v0 = w * h1[r * 8 + offin];
  float v1 = w * h1[r * 8 + offin + 1];
  atomicAdd(&h1[c * 8 + offout],     v0);
  atomicAdd(&h1[c * 8 + offout + 1], v1);
}

// X1 = relu(h1(Nx8) @ W1cat(8x64) + b1)  ->  Hcat slice 0; zero slices 1..3
__global__ void k_l1_combine(const float* __restrict__ h1, const float* __restrict__ W1,
                             const float* __restrict__ b1, float* __restrict__ H, ll n) {
  ll t = (ll)blockIdx.x * blockDim.x + threadIdx.x;
  if (t >= n * 64) return;
  ll node = t >> 6; int j = (int)(t & 63);
  const float* hn = h1 + node * 8;
  float acc = b1[j];
#pragma unroll
  for (int kk = 0; kk < 8; ++kk) acc += hn[kk] * W1[kk * 64 + j];
  acc = acc > 0.f ? acc : 0.f;
  ll base = node * 256;
  H[base + j] = acc;
  H[base + 64 + j] = 0.f;
  H[base + 128 + j] = 0.f;
  H[base + 192 + j] = 0.f;
}

// ---------------- 64-wide propagation hop ----------------
// H[col][offout+f] += norm * H[row][offin+f], float4 per thread (16 threads/edge)
__global__ void k_prop64(const ll* __restrict__ row, const ll* __restrict__ col,
                         const float* __restrict__ norm, float* __restrict__ H,
                         int offin, int offout, ll E) {
  ll t = (ll)blockIdx.x * blockDim.x + threadIdx.x;
  ll total = E * 16;
  if (t >= total) return;
  ll e = t >> 4;
  int q = (int)(t & 15);
  ll r = row[e], c = col[e];
  float w = norm[e];
  const float* src = H + r * 256 + offin + q * 4;
  float* dst = H + c * 256 + offout + q * 4;
  v4f v = *(const v4f*)src;
  atomicAdd(dst + 0, w * v.x);
  atomicAdd(dst + 1, w * v.y);
  atomicAdd(dst + 2, w * v.z);
  atomicAdd(dst + 3, w * v.w);
}

// ---------------- layer 2 GEMM: WMMA f32 16x16x4 ----------------
// In-place: X2 = relu(Hcat(Nx256) @ W(256x64) + b); write slice0, zero slices 1..3.
// 128 threads = 4 waves; block owns 16 rows; wave w owns 16 output cols.
__global__ __launch_bounds__(128)
void k_gemm_wmma(float* __restrict__ H, const float* __restrict__ W,
                 const float* __restrict__ bias, ll n) {
  __shared__ float As[16][260];           // pad 260: bank-conflict-free column reads
  const int m0 = blockIdx.x * 16;
  const int tid = threadIdx.x;

  // cooperative stage of the 16x256 A tile (float4)
  for (int i = tid; i < 16 * 64; i += 128) {
    int r = i >> 6, c4 = i & 63;
    ll gr = m0 + r;
    v4f v = {0.f, 0.f, 0.f, 0.f};
    if (gr < n) v = *(const v4f*)(H + gr * 256 + c4 * 4);
    *(v4f*)(&As[r][c4 * 4]) = v;
  }
  __syncthreads();

  const int wave = tid >> 5;
  const int lane = tid & 31;
  const int half = lane >> 4;             // K sub-pair selector per ISA A/B layout
  const int lm   = lane & 15;
  const int n0   = wave * 16;

  v8f acc = {};
  for (int kk = 0; kk < 256; kk += 4) {
    const int ka = kk + 2 * half;
    v2f a, b;
    a.x = As[lm][ka];                     // A: 16x4, lanes0-15 K={0,1}, lanes16-31 K={2,3}
    a.y = As[lm][ka + 1];
    b.x = W[(ll)ka * 64 + n0 + lm];       // B: 4x16, row-striped across lanes
    b.y = W[(ll)(ka + 1) * 64 + n0 + lm];
    acc = __builtin_amdgcn_wmma_f32_16x16x4_f32(
        /*neg_a=*/false, a, /*neg_b=*/false, b,
        /*c_mod=*/(short)0, acc, /*reuse_a=*/false, /*reuse_b=*/false);
  }

  const float bj = bias[n0 + lm];
#pragma unroll
  for (int v = 0; v < 8; ++v) {           // D: VGPR v -> row v (+8 for upper half-lanes)
    ll gr = (ll)m0 + v + 8 * half;
    if (gr < n) {
      float val = acc[v] + bj;
      val = val > 0.f ? val : 0.f;
      ll base = gr * 256;
      H[base + n0 + lm]        = val;     // slice 0 = X2
      H[base + 64  + n0 + lm]  = 0.f;     // prime hop slices for layer 3
      H[base + 128 + n0 + lm]  = 0.f;
      H[base + 192 + n0 + lm]  = 0.f;
    }
  }
}

// ---------------- layer 3 reduce + pool ----------------
// per node: dot(Hcat[n][0:256], W3flat[0:256]) + b3 -> LDS pool -> global pool
__global__ __launch_bounds__(256)
void k_l3_pool(const float* __restrict__ H, const float* __restrict__ W3,
               const float* __restrict__ b3, const ll* __restrict__ batch,
               float* __restrict__ pool, ll n, int ng) {
  __shared__ float sp[64];
  if (threadIdx.x < 64) sp[threadIdx.x] = 0.f;
  __syncthreads();

  ll node = (ll)blockIdx.x * blockDim.x + threadIdx.x;
  if (node < n) {
    const float* hn = H + node * 256;
    float acc = 0.f;
#pragma unroll 8
    for (int i = 0; i < 64; ++i) {
      v4f a = *(const v4f*)(hn + i * 4);
      v4f w = *(const v4f*)(W3 + i * 4);
      acc += a.x * w.x + a.y * w.y + a.z * w.z + a.w * w.w;
    }
    acc += b3[0];
    int g = (int)batch[node];
    if (g >= 0 && g < 64) atomicAdd(&sp[g], acc);   // ds_add_f32
  }
  __syncthreads();
  if (threadIdx.x < 64 && (int)threadIdx.x < ng && sp[threadIdx.x] != 0.f)
    atomicAdd(&pool[threadIdx.x], sp[threadIdx.x]); // global_atomic_add_f32
}

__global__ void k_sigmoid(const float* __restrict__ pool, float* __restrict__ out, int ng) {
  int g = blockIdx.x * blockDim.x + threadIdx.x;
  if (g < ng) out[g] = 1.f / (1.f + expf(-pool[g]));
}

// ---------------- host launcher ----------------
extern "C" void kernel_launch(void* const* d_in, const int* in_sizes, int n_in,
                              void* d_out, int out_size, void* d_ws, size_t ws_size,
                              hipStream_t stream) {
  const float* x   = (const float*)d_in[0];
  const ll*    ei  = (const ll*)d_in[1];     // int64 (2,E): [row | col]
  const float* ew  = (const float*)d_in[2];
  const ll*    bat = (const ll*)d_in[3];
  const float* W1  = (const float*)d_in[4];  // (4,2,64)  == 8x64 concat
  const float* b1  = (const float*)d_in[5];
  const float* W2  = (const float*)d_in[6];  // (4,64,64) == 256x64 concat
  const float* b2  = (const float*)d_in[7];
  const float* W3  = (const float*)d_in[8];  // (4,64,1)  == 256 concat vector
  const float* b3  = (const float*)d_in[9];

  const ll N  = (ll)in_sizes[0] / 2;
  const ll E  = (ll)in_sizes[2];
  const int NG = out_size;                   // 64 graphs x F_OUT=1
  const ll* row = ei;
  const ll* col = ei + E;

  // workspace layout (16-float aligned slabs)
  auto align16 = [](ll v) { return (v + 15) & ~(ll)15; };
  float* ws   = (float*)d_ws;
  float* deg  = ws;                                   // N  (becomes dinv in-place)
  float* norm = deg  + align16(N);                    // E
  float* h1   = norm + align16(E);                    // N*8
  float* Hc   = h1   + align16(N * 8);                // N*256
  float* pool = Hc   + align16(N * 256);              // NG

  const int T = 256;
  auto blk = [&](ll work) { return (unsigned)((work + T - 1) / T); };

  // degree -> dinv -> edge norm
  k_zero<<<blk(N), T, 0, stream>>>(deg, N);
  k_zero<<<blk(NG), T, 0, stream>>>(pool, NG);
  k_deg<<<blk(E), T, 0, stream>>>(col, ew, deg, E);
  k_dinv<<<blk(N), T, 0, stream>>>(deg, N);
  k_norm<<<blk(E), T, 0, stream>>>(row, col, ew, deg, norm, E);

  // ---- layer 1 (2-wide hops, scalar combine) ----
  k_init_h1<<<blk(N), T, 0, stream>>>(x, h1, N);
  k_prop2<<<blk(E), T, 0, stream>>>(row, col, norm, h1, 0, 2, E);
  k_prop2<<<blk(E), T, 0, stream>>>(row, col, norm, h1, 2, 4, E);
  k_prop2<<<blk(E), T, 0, stream>>>(row, col, norm, h1, 4, 6, E);
  k_l1_combine<<<blk(N * 64), T, 0, stream>>>(h1, W1, b1, Hc, N);

  // ---- layer 2: 3 hops of 64-wide propagation, then fused WMMA GEMM ----
  k_prop64<<<blk(E * 16), T, 0, stream>>>(row, col, norm, Hc, 0,   64,  E);
  k_prop64<<<blk(E * 16), T, 0, stream>>>(row, col, norm, Hc, 64,  128, E);
  k_prop64<<<blk(E * 16), T, 0, stream>>>(row, col, norm, Hc, 128, 192, E);
  k_gemm_wmma<<<(unsigned)((N + 15) / 16), 128, 0, stream>>>(Hc, W2, b2, N);

  // ---- layer 3: 3 more hops, reduce to scalar per node, pool, sigmoid ----
  k_prop64<<<blk(E * 16), T, 0, stream>>>(row, col, norm, Hc, 0,   64,  E);
  k_prop64<<<blk(E * 16), T, 0, stream>>>(row, col, norm, Hc, 64,  128, E);
  k_prop64<<<blk(E * 16), T, 0, stream>>>(row, col, norm, Hc, 128, 192, E);
  k_l3_pool<<<blk(N), T, 0, stream>>>(Hc, W3, b3, bat, pool, N, NG);
  k_sigmoid<<<blk(NG), T, 0, stream>>>(pool, (float*)d_out, NG);
}